// HeteroGNN_790273982767
// MI455X (gfx1250) — compile-verified
//
#include <hip/hip_runtime.h>

typedef __attribute__((ext_vector_type(2))) float v2f;
typedef __attribute__((ext_vector_type(8))) float v8f;

#define NNODES   50000
#define CIN      128
#define COUT     128
#define NEDGE    800000
#define ROWTILES 3125            // 50000 / 16 (exact)
#define TPW      4               // row tiles per wave
#define TPB      32              // row tiles per block (8 waves * TPW)
#define YBLOCKS  98              // ceil(3125 / 32)
#define BSTRIDE  132             // LDS row stride, K=128 tiles: bank=(4*ln+kk)%64, conflict-free
#define BSTRIDE3 260             // LDS row stride, K=256 tile:  260%64=4 -> same property

__device__ __forceinline__ float sigmoidf_(float x) {
    return 1.0f / (1.0f + __expf(-x));
}
// Branch-free tanh (v_exp_f32 + fast divide), avoids ocml exec-mask branches.
__device__ __forceinline__ float fast_tanhf_(float x) {
    float ax = __builtin_fabsf(x);
    float t  = __expf(-2.0f * ax);
    float r  = __fdividef(1.0f - t, 1.0f + t);
    return __builtin_copysignf(r, x);
}

// ---------------------------------------------------------------------------
// Kernel 1: LSTM(seq=1, c0=0) message: msg = sigmoid(o)*tanh(sigmoid(i)*tanh(g))
// gates = x @ W_ih^T + b ; W rows [i:0..127, f:128..255, g:256..383, o:384..511];
// f-gate dead. Block = one 16-col output slice x 32 row-tiles. The i/g/o W-tile
// (3 x 16 x 128) is staged in LDS ONCE per block and shared by all 8 waves
// (B reuse over 512 rows -> W traffic ~19 MB instead of 614 MB). Each wave
// keeps 12 fp32-WMMA accumulators (4 row tiles x {i,g,o}).
// ---------------------------------------------------------------------------
__global__ __launch_bounds__(256) void lstm_msg_kernel(
    const float* __restrict__ x,     // [N, 128]
    const float* __restrict__ W,     // [512, 128] row-major
    const float* __restrict__ bias,  // [512]
    float* __restrict__ msg)         // [N, 128]
{
    __shared__ float sB[3][16][BSTRIDE];

    const int wave = threadIdx.x >> 5;
    const int lane = threadIdx.x & 31;
    const int half = lane >> 4;          // K pair {k0,k0+1} vs {k0+2,k0+3}
    const int ln   = lane & 15;
    const int col0 = blockIdx.x * 16;    // output col slice (per gate)

    // Stage B: W rows [gate + col0 .. +16), K = 0..127, for gates i,g,o.
    {
        const int r = threadIdx.x >> 4;
        const int c = (threadIdx.x & 15) * 8;
        const int gateoff[3] = {0, 2 * COUT, 3 * COUT};
#pragma unroll
        for (int g = 0; g < 3; ++g) {
            const float* wr = W + (gateoff[g] + col0 + r) * CIN + c;
            *(float4*)(&sB[g][r][c])     = *(const float4*)(wr);
            *(float4*)(&sB[g][r][c + 4]) = *(const float4*)(wr + 4);
        }
    }
    __syncthreads();

    // Row-tile bases for this wave (clamped tail: duplicate identical writes).
    const float* ax[TPW];
    int rowb[TPW];
#pragma unroll
    for (int tt = 0; tt < TPW; ++tt) {
        int tile = blockIdx.y * TPB + wave * TPW + tt;
        tile = tile < ROWTILES ? tile : ROWTILES - 1;
        rowb[tt] = tile * 16;
        ax[tt] = x + (rowb[tt] + ln) * CIN;
    }

    v8f ai[TPW] = {}; v8f ag[TPW] = {}; v8f ao[TPW] = {};

    for (int k0 = 0; k0 < CIN; k0 += 4) {
        const int kk = k0 + 2 * half;
        v2f bi = *(const v2f*)(&sB[0][ln][kk]);
        v2f bg = *(const v2f*)(&sB[1][ln][kk]);
        v2f bo = *(const v2f*)(&sB[2][ln][kk]);
#pragma unroll
        for (int tt = 0; tt < TPW; ++tt) {
            v2f a = *(const v2f*)(ax[tt] + kk);
            ai[tt] = __builtin_amdgcn_wmma_f32_16x16x4_f32(false, a, false, bi, (short)0, ai[tt], false, false);
            ag[tt] = __builtin_amdgcn_wmma_f32_16x16x4_f32(false, a, false, bg, (short)0, ag[tt], false, false);
            ao[tt] = __builtin_amdgcn_wmma_f32_16x16x4_f32(false, a, false, bo, (short)0, ao[tt], false, false);
        }
    }

    const float bI = bias[0 * COUT + col0 + ln];
    const float bG = bias[2 * COUT + col0 + ln];
    const float bO = bias[3 * COUT + col0 + ln];

#pragma unroll
    for (int tt = 0; tt < TPW; ++tt) {
#pragma unroll
        for (int v = 0; v < 8; ++v) {
            const int m = v + 8 * half;      // C/D layout: lanes>=16 hold M+8
            float c = sigmoidf_(ai[tt][v] + bI) * fast_tanhf_(ag[tt][v] + bG);
            float h = sigmoidf_(ao[tt][v] + bO) * fast_tanhf_(c);
            msg[(rowb[tt] + m) * COUT + col0 + ln] = h;
        }
    }
}

// ---------------------------------------------------------------------------
// Kernel 2: edge scatter-add. One wave per edge: lane handles 4 channels.
// msg/aggr (25.6 MB each) are L2-resident vs 192 MB L2 -> L2-atomic bound.
// ---------------------------------------------------------------------------
__global__ __launch_bounds__(256) void scatter_add_kernel(
    const float* __restrict__ msg,
    const int* __restrict__ src,
    const int* __restrict__ dst,
    float* __restrict__ aggr,
    int nedges)
{
    const int gid  = blockIdx.x * blockDim.x + threadIdx.x;
    const int e    = gid >> 5;
    const int lane = gid & 31;
    if (e >= nedges) return;
    const int s = src[e];
    const int d = dst[e];
    const float4 m = *(const float4*)(msg + s * COUT + lane * 4);
    float* ap = aggr + d * COUT + lane * 4;
    unsafeAtomicAdd(ap + 0, m.x);
    unsafeAtomicAdd(ap + 1, m.y);
    unsafeAtomicAdd(ap + 2, m.z);
    unsafeAtomicAdd(ap + 3, m.w);
}

// ---------------------------------------------------------------------------
// Kernel 3: out = relu(concat(x_dst, aggr) @ W_lin^T + b_lin)
// Same blocking: one 16-col output slice per block, W_lin tile (16 x 256)
// staged in LDS once, 4 row tiles per wave. K split into two 128-chunks
// (x_dst then aggr) as two clean loops.
// ---------------------------------------------------------------------------
__global__ __launch_bounds__(256) void out_gemm_kernel(
    const float* __restrict__ xdst,  // [N, 128]
    const float* __restrict__ aggr,  // [N, 128]
    const float* __restrict__ Wl,    // [128, 256] row-major
    const float* __restrict__ bl,    // [128]
    float* __restrict__ out)         // [N, 128]
{
    __shared__ float sB[16][BSTRIDE3];

    const int wave = threadIdx.x >> 5;
    const int lane = threadIdx.x & 31;
    const int half = lane >> 4;
    const int ln   = lane & 15;
    const int col0 = blockIdx.x * 16;

    // Stage B: W_lin rows [col0 .. col0+16), K = 0..255.
    {
        const int r = threadIdx.x >> 4;
        const int c = (threadIdx.x & 15) * 16;
        const float* wr = Wl + (col0 + r) * (CIN + COUT) + c;
#pragma unroll
        for (int q = 0; q < 4; ++q)
            *(float4*)(&sB[r][c + 4 * q]) = *(const float4*)(wr + 4 * q);
    }
    __syncthreads();

    const float* axd[TPW];
    const float* agg[TPW];
    int rowb[TPW];
#pragma unroll
    for (int tt = 0; tt < TPW; ++tt) {
        int tile = blockIdx.y * TPB + wave * TPW + tt;
        tile = tile < ROWTILES ? tile : ROWTILES - 1;
        rowb[tt] = tile * 16;
        axd[tt] = xdst + (rowb[tt] + ln) * CIN;
        agg[tt] = aggr + (rowb[tt] + ln) * CIN;
    }

    v8f acc[TPW] = {};

    for (int k0 = 0; k0 < CIN; k0 += 4) {            // first 128 K: x_dst
        const int kk = k0 + 2 * half;
        v2f b = *(const v2f*)(&sB[ln][kk]);
#pragma unroll
        for (int tt = 0; tt < TPW; ++tt) {
            v2f a = *(const v2f*)(axd[tt] + kk);
            acc[tt] = __builtin_amdgcn_wmma_f32_16x16x4_f32(false, a, false, b, (short)0, acc[tt], false, false);
        }
    }
    for (int k0 = 0; k0 < COUT; k0 += 4) {           // next 128 K: aggr
        const int kk = k0 + 2 * half;
        v2f b = *(const v2f*)(&sB[ln][CIN + kk]);
#pragma unroll
        for (int tt = 0; tt < TPW; ++tt) {
            v2f a = *(const v2f*)(agg[tt] + kk);
            acc[tt] = __builtin_amdgcn_wmma_f32_16x16x4_f32(false, a, false, b, (short)0, acc[tt], false, false);
        }
    }

    const float bb = bl[col0 + ln];
#pragma unroll
    for (int tt = 0; tt < TPW; ++tt) {
#pragma unroll
        for (int v = 0; v < 8; ++v) {
            const int m = v + 8 * half;
            out[(rowb[tt] + m) * COUT + col0 + ln] = fmaxf(acc[tt][v] + bb, 0.0f);
        }
    }
}

// ---------------------------------------------------------------------------
extern "C" void kernel_launch(void* const* d_in, const int* in_sizes, int n_in,
                              void* d_out, int out_size, void* d_ws, size_t ws_size,
                              hipStream_t stream) {
    const float* x_a     = (const float*)d_in[0];
    const float* x_b     = (const float*)d_in[1];
    const int*   e_ab    = (const int*)d_in[2];   // [2, E]: row 0 = src, row 1 = dst
    const int*   e_ba    = (const int*)d_in[3];
    const float* Wih_ab  = (const float*)d_in[4];
    const float* b_ab    = (const float*)d_in[5];
    const float* Wlin_ab = (const float*)d_in[6];
    const float* blin_ab = (const float*)d_in[7];
    const float* Wih_ba  = (const float*)d_in[8];
    const float* b_ba    = (const float*)d_in[9];
    const float* Wlin_ba = (const float*)d_in[10];
    const float* blin_ba = (const float*)d_in[11];

    float* out_a = (float*)d_out;                          // stack index 0
    float* out_b = (float*)d_out + (size_t)NNODES * COUT;  // stack index 1

    float* msg  = (float*)d_ws;                            // 25.6 MB
    float* aggr = msg + (size_t)NNODES * COUT;             // 25.6 MB

    const dim3 blk(256);
    const dim3 gemmGrid(COUT / 16, YBLOCKS);               // 8 x 98
    const int scatBlocks = (NEDGE * 32 + 255) / 256;
    const size_t aggrBytes = (size_t)NNODES * COUT * sizeof(float);

    // -------- direction a -> b (updates type-b nodes) --------
    lstm_msg_kernel<<<gemmGrid, blk, 0, stream>>>(x_a, Wih_ab, b_ab, msg);
    hipMemsetAsync(aggr, 0, aggrBytes, stream);
    scatter_add_kernel<<<scatBlocks, blk, 0, stream>>>(msg, e_ab, e_ab + NEDGE, aggr, NEDGE);
    out_gemm_kernel<<<gemmGrid, blk, 0, stream>>>(x_b, aggr, Wlin_ab, blin_ab, out_b);

    // -------- direction b -> a (updates type-a nodes) --------
    lstm_msg_kernel<<<gemmGrid, blk, 0, stream>>>(x_b, Wih_ba, b_ba, msg);
    hipMemsetAsync(aggr, 0, aggrBytes, stream);
    scatter_add_kernel<<<scatBlocks, blk, 0, stream>>>(msg, e_ba, e_ba + NEDGE, aggr, NEDGE);
    out_gemm_kernel<<<gemmGrid, blk, 0, stream>>>(x_a, aggr, Wlin_ba, blin_ba, out_a);
}